// ModelNew_33921651704505
// MI455X (gfx1250) — compile-verified
//
#include <hip/hip_runtime.h>
#include <hip/hip_bf16.h>
#include <stdint.h>

typedef __attribute__((ext_vector_type(2))) float v2f;
typedef __attribute__((ext_vector_type(8))) float v8f;

// ---------------- problem constants ----------------
constexpr int C_IN = 3,  H0 = 224;
constexpr int C1   = 64, HC1 = 220, HP1 = 110;   // conv1 out / pool1 out spatial
constexpr int C2   = 128, HC2 = 106, HP2 = 53;   // conv2 out / pool2 out spatial
constexpr int KT1  = C_IN * 25;                  // 75
constexpr int KT2  = C1 * 25;                    // 1600
constexpr int M1   = HC1 * HC1;                  // 48400 output pixels conv1
constexpr int M2   = HC2 * HC2;                  // 11236 output pixels conv2
constexpr int FLAT = C2 * HP2 * HP2;             // 359552
constexpr int FC1N = 1024, FC2N = 512, NCLS = 100;

constexpr int KC = 64;                            // K-chunk staged in LDS per step
constexpr int KT1_PAD = ((KT1 + KC - 1) / KC) * KC;   // 128 (wt1 region padded so async
constexpr int KT2_PAD = ((KT2 + KC - 1) / KC) * KC;   // 1600   loads never go OOB)

// ---------------- workspace layout (floats) ----------------
constexpr size_t OFF_C1OUT = 0;                                  // [M1][C1] pixel-major
constexpr size_t OFF_P1    = OFF_C1OUT + (size_t)M1 * C1;        // [C1][110][110] NCHW
constexpr size_t OFF_C2OUT = OFF_P1    + (size_t)C1 * HP1 * HP1; // [M2][C2] pixel-major
constexpr size_t OFF_P2    = OFF_C2OUT + (size_t)M2 * C2;        // [C2][53][53] (= flatten)
constexpr size_t OFF_WT1   = OFF_P2    + (size_t)FLAT;           // [KT1_PAD][C1]
constexpr size_t OFF_WT2   = OFF_WT1   + (size_t)KT1_PAD * C1;   // [KT2_PAD][C2]
constexpr size_t OFF_FC1   = OFF_WT2   + (size_t)KT2_PAD * C2;   // [1024]
constexpr size_t OFF_FC2   = OFF_FC1   + FC1N;                   // [512]

// ---------------- weight transpose: OIHW [n][k] -> [k][n] ----------------
__global__ __launch_bounds__(256) void wtrans_kernel(const float* __restrict__ w,
                                                     float* __restrict__ wt,
                                                     int Cout, int Ktot) {
    int idx = blockIdx.x * blockDim.x + threadIdx.x;
    if (idx >= Cout * Ktot) return;
    int n = idx / Ktot;
    int k = idx - n * Ktot;
    wt[k * Cout + n] = w[idx];
}

// ------- fused conv (implicit GEMM, f32 WMMA, async-LDS staging) + bias + ReLU -------
// D[M x Cout] = im2col(A)[M x Ktot] * Wt[Ktot x Cout]; out pixel-major [m][n].
// One wave computes a 32(M) x 16(N) tile: 2 WMMAs per K-step sharing one B fragment.
__global__ __launch_bounds__(32) void conv_wmma_kernel(const float* __restrict__ in,
                                                       const float* __restrict__ wt,
                                                       const float* __restrict__ bias,
                                                       float* __restrict__ out,
                                                       int Hin, int Win,
                                                       int Cout, int Wout,
                                                       int Ktot, int M) {
    extern __shared__ char smem[];
    float* lb  = reinterpret_cast<float*>(smem);                    // [KC][16] B tile
    int*   off = reinterpret_cast<int*>(smem + KC * 16 * sizeof(float)); // [Kpad] im2col offsets

    const int lane = threadIdx.x;          // wave32
    const int hv   = lane >> 4;            // lane half: A/B frags hold K 0,1 | 2,3
    const int lr   = lane & 15;

    const int m0 = blockIdx.x * 32;
    const int n0 = blockIdx.y * 16;
    const int n  = n0 + lr;

    const int Kpad = ((Ktot + KC - 1) / KC) * KC;

    // --- precompute per-k im2col offsets (padding -> 0, stays in-bounds) ---
    for (int k = lane; k < Kpad; k += 32) {
        int o = 0;
        if (k < Ktot) {
            const int c   = k / 25;
            const int rem = k - c * 25;
            const int r   = rem / 5;
            const int s   = rem - r * 5;
            o = (c * Hin + r) * Win + s;
        }
        off[k] = o;
    }

    // --- per-lane A-row bases (two M sub-tiles) ---
    const int mA0 = m0 + lr, mA1 = m0 + 16 + lr;
    const int oh0 = (mA0 < M) ? (mA0 / Wout) : 0;
    const int ow0 = (mA0 < M) ? (mA0 - oh0 * Wout) : 0;
    const int oh1 = (mA1 < M) ? (mA1 / Wout) : 0;
    const int ow1 = (mA1 < M) ? (mA1 - oh1 * Wout) : 0;
    const float* __restrict__ inA0 = in + (size_t)(oh0 * Win + ow0);
    const float* __restrict__ inA1 = in + (size_t)(oh1 * Win + ow1);

    const unsigned lb_lds = (unsigned)(uintptr_t)lb;  // LDS byte offset (aperture low bits)

    v8f acc0 = {0.f, 0.f, 0.f, 0.f, 0.f, 0.f, 0.f, 0.f};
    v8f acc1 = {0.f, 0.f, 0.f, 0.f, 0.f, 0.f, 0.f, 0.f};

    __syncthreads();

    for (int kbase = 0; kbase < Ktot; kbase += KC) {
        __syncthreads();
        // --- async-stage B chunk [KC][16] global -> LDS (Tensor/async data path) ---
#pragma unroll
        for (int it = 0; it < (KC * 16 / 4) / 32; ++it) {   // 8 x b128 per lane-set
            const int g    = it * 32 + lane;                // float4 group id
            const int krow = g >> 2;
            const int col4 = (g & 3) << 2;
            const float* gp = wt + (size_t)(kbase + krow) * Cout + (n0 + col4);
            const unsigned lo = lb_lds + (unsigned)(((krow << 4) + col4) * 4);
            asm volatile("global_load_async_to_lds_b128 %0, %1, off"
                         :: "v"(lo), "v"(gp) : "memory");
        }
        asm volatile("s_wait_asynccnt 0x0" ::: "memory");
        // zero padded K rows (only the final partial chunk takes this branch)
        if (kbase + KC > Ktot) {
            for (int i = lane; i < KC * 16; i += 32)
                if (kbase + (i >> 4) >= Ktot) lb[i] = 0.f;
        }
        __syncthreads();

        // --- inner product over the chunk: branch-free, 2 WMMAs per K-step ---
#pragma unroll 4
        for (int kk = 0; kk < KC; kk += 4) {
            const int ka = kk + (hv << 1);
            const int kg = kbase + ka;
            const int2 o2 = *reinterpret_cast<const int2*>(off + kg);  // ds_load_b64
            v2f bf, a0, a1;
            bf[0] = lb[(ka << 4) + lr];
            bf[1] = lb[(ka << 4) + 16 + lr];
            a0[0] = inA0[o2.x];  a0[1] = inA0[o2.y];
            a1[0] = inA1[o2.x];  a1[1] = inA1[o2.y];
            acc0 = __builtin_amdgcn_wmma_f32_16x16x4_f32(false, a0, false, bf,
                                                         (short)0, acc0, false, false);
            acc1 = __builtin_amdgcn_wmma_f32_16x16x4_f32(false, a1, false, bf,
                                                         (short)0, acc1, false, false);
        }
    }

    // --- epilogue: bias + ReLU, pixel-major store (lane-contiguous) ---
    const float bn = bias[n];
#pragma unroll
    for (int t = 0; t < 2; ++t) {
        const v8f acc = t ? acc1 : acc0;
#pragma unroll
        for (int r = 0; r < 8; ++r) {
            const int mo = m0 + t * 16 + (hv << 3) + r;
            if (mo < M) {
                const float v = acc[r] + bn;
                out[(size_t)mo * Cout + n] = fmaxf(v, 0.f);
            }
        }
    }
}

// ---------------- 2x2/s2 max-pool, [m][n] pixel-major -> NCHW [c][ph][pw] ---
__global__ __launch_bounds__(256) void pool_kernel(const float* __restrict__ conv,
                                                   float* __restrict__ pooled,
                                                   int C, int W, int PH, int PW) {
    int idx = blockIdx.x * blockDim.x + threadIdx.x;
    if (idx >= C * PH * PW) return;
    int c   = idx / (PH * PW);
    int rem = idx - c * (PH * PW);
    int ph  = rem / PW;
    int pw  = rem - ph * PW;
    size_t m00 = (size_t)(2 * ph) * W + 2 * pw;
    float a = conv[(m00)         * C + c];
    float b = conv[(m00 + 1)     * C + c];
    float d = conv[(m00 + W)     * C + c];
    float e = conv[(m00 + W + 1) * C + c];
    pooled[idx] = fmaxf(fmaxf(a, b), fmaxf(d, e));
}

// ---------------- bandwidth-bound GEMV (+bias, optional ReLU) ---------------
__global__ __launch_bounds__(256) void gemv_kernel(const float* __restrict__ W,
                                                   const float* __restrict__ x,
                                                   const float* __restrict__ bias,
                                                   float* __restrict__ y,
                                                   int K4, int doRelu) {
    __shared__ float red[256];
    const int row = blockIdx.x;
    const float4* __restrict__ Wr = reinterpret_cast<const float4*>(W) + (size_t)row * K4;
    const float4* __restrict__ xv = reinterpret_cast<const float4*>(x);

    float p = 0.f;
    for (int i = threadIdx.x; i < K4; i += blockDim.x) {
        __builtin_prefetch(Wr + i + blockDim.x, 0, 0);   // global_prefetch_b8
        float4 w4 = Wr[i];
        float4 x4 = xv[i];
        p = fmaf(w4.x, x4.x, p);
        p = fmaf(w4.y, x4.y, p);
        p = fmaf(w4.z, x4.z, p);
        p = fmaf(w4.w, x4.w, p);
    }
    red[threadIdx.x] = p;
    __syncthreads();
    for (int s = 128; s > 0; s >>= 1) {
        if ((int)threadIdx.x < s) red[threadIdx.x] += red[threadIdx.x + s];
        __syncthreads();
    }
    if (threadIdx.x == 0) {
        float v = red[0] + bias[row];
        y[row] = doRelu ? fmaxf(v, 0.f) : v;
    }
}

// ---------------- launcher ----------------
extern "C" void kernel_launch(void* const* d_in, const int* in_sizes, int n_in,
                              void* d_out, int out_size, void* d_ws, size_t ws_size,
                              hipStream_t stream) {
    (void)in_sizes; (void)n_in; (void)out_size; (void)ws_size;

    const float* x       = (const float*)d_in[0];
    const float* conv1_w = (const float*)d_in[1];
    const float* conv1_b = (const float*)d_in[2];
    const float* conv2_w = (const float*)d_in[3];
    const float* conv2_b = (const float*)d_in[4];
    const float* fc1_w   = (const float*)d_in[5];
    const float* fc1_b   = (const float*)d_in[6];
    const float* fc2_w   = (const float*)d_in[7];
    const float* fc2_b   = (const float*)d_in[8];
    const float* fc3_w   = (const float*)d_in[9];
    const float* fc3_b   = (const float*)d_in[10];
    float* out = (float*)d_out;
    float* ws  = (float*)d_ws;

    float* c1out = ws + OFF_C1OUT;
    float* p1    = ws + OFF_P1;
    float* c2out = ws + OFF_C2OUT;
    float* p2    = ws + OFF_P2;
    float* wt1   = ws + OFF_WT1;
    float* wt2   = ws + OFF_WT2;
    float* fc1o  = ws + OFF_FC1;
    float* fc2o  = ws + OFF_FC2;

    // 1) weight transposes OIHW -> [K][Cout]
    wtrans_kernel<<<(C1 * KT1 + 255) / 256, 256, 0, stream>>>(conv1_w, wt1, C1, KT1);
    wtrans_kernel<<<(C2 * KT2 + 255) / 256, 256, 0, stream>>>(conv2_w, wt2, C2, KT2);

    // 2) conv1 (WMMA implicit GEMM) + bias + ReLU
    {
        dim3 grid((M1 + 31) / 32, C1 / 16);
        size_t shmem = KC * 16 * sizeof(float) + KT1_PAD * sizeof(int);
        conv_wmma_kernel<<<grid, 32, shmem, stream>>>(x, wt1, conv1_b, c1out,
                                                      H0, H0, C1, HC1, KT1, M1);
    }
    // 3) pool1 -> NCHW
    pool_kernel<<<(C1 * HP1 * HP1 + 255) / 256, 256, 0, stream>>>(c1out, p1, C1, HC1, HP1, HP1);

    // 4) conv2 (WMMA implicit GEMM) + bias + ReLU
    {
        dim3 grid((M2 + 31) / 32, C2 / 16);
        size_t shmem = KC * 16 * sizeof(float) + KT2_PAD * sizeof(int);
        conv_wmma_kernel<<<grid, 32, shmem, stream>>>(p1, wt2, conv2_b, c2out,
                                                      HP1, HP1, C2, HC2, KT2, M2);
    }
    // 5) pool2 -> NCHW (== flatten order)
    pool_kernel<<<(C2 * HP2 * HP2 + 255) / 256, 256, 0, stream>>>(c2out, p2, C2, HC2, HP2, HP2);

    // 6) fc1: 1024 x 359552 GEMV — HBM-bandwidth bound (1.47 GB weights, ~63 us floor)
    gemv_kernel<<<FC1N, 256, 0, stream>>>(fc1_w, p2, fc1_b, fc1o, FLAT / 4, 1);
    // 7) fc2: 512 x 1024
    gemv_kernel<<<FC2N, 256, 0, stream>>>(fc2_w, fc1o, fc2_b, fc2o, FC1N / 4, 1);
    // 8) fc3: 100 x 512 (no ReLU) -> d_out
    gemv_kernel<<<NCLS, 256, 0, stream>>>(fc3_w, fc2o, fc3_b, out, FC2N / 4, 0);
}